// GDN_20607253087063
// MI455X (gfx1250) — compile-verified
//
#include <hip/hip_runtime.h>
#include <hip/hip_bf16.h>
#include <math.h>

// Sizes fixed by the reference
#define BB    64
#define NN    1024
#define FIN   10
#define DIM   64
#define TOPK  20
#define NODES (BB * NN)     // 65536
#define EPS   1e-5f
#define NEG_SLOPE 0.2f

typedef float v2f __attribute__((ext_vector_type(2)));
typedef float v8f __attribute__((ext_vector_type(8)));

__device__ __forceinline__ float wave_sum(float v) {
    #pragma unroll
    for (int m = 16; m > 0; m >>= 1) v += __shfl_xor(v, m, 32);
    return v;
}
__device__ __forceinline__ float wave_max(float v) {
    #pragma unroll
    for (int m = 16; m > 0; m >>= 1) v = fmaxf(v, __shfl_xor(v, m, 32));
    return v;
}

// ---------------- Kernel 1: row-normalize emb -> wn (so wn@wn^T == cosine) ---
__global__ void k_normalize(const float* __restrict__ emb, float* __restrict__ wn) {
    int n = blockIdx.x;            // node 0..1023
    int l = threadIdx.x;           // lane 0..31
    float a = emb[n * DIM + l];
    float b = emb[n * DIM + l + 32];
    float s = wave_sum(a * a + b * b);
    float inv = 1.0f / sqrtf(s);
    wn[n * DIM + l]      = a * inv;
    wn[n * DIM + l + 32] = b * inv;
}

// ---------------- Kernel 2: cos = wn @ wn^T via V_WMMA_F32_16X16X4_F32 -------
// One 16x16 output tile per wave; 64x64 tiles total; 16 k-steps of K=4.
__global__ void k_cos(const float* __restrict__ wn, float* __restrict__ cosm) {
    int lane = threadIdx.x & 31;
    int wave = threadIdx.x >> 5;
    int tile = blockIdx.x * 8 + wave;          // 0..4095
    int tm = tile >> 6, tn = tile & 63;
    int r16   = lane & 15;
    int khalf = (lane >> 4) * 2;               // lanes 0-15 -> K+{0,1}; 16-31 -> K+{2,3}
    const float* arow = wn + (tm * 16 + r16) * DIM + khalf;
    const float* brow = wn + (tn * 16 + r16) * DIM + khalf;   // B[k][n] = wn[n][k]
    v8f c = {};
    #pragma unroll
    for (int k = 0; k < DIM; k += 4) {
        v2f a, b;
        a[0] = arow[k]; a[1] = arow[k + 1];
        b[0] = brow[k]; b[1] = brow[k + 1];
        c = __builtin_amdgcn_wmma_f32_16x16x4_f32(false, a, false, b, (short)0, c,
                                                  false, false);
    }
    int col    = tn * 16 + r16;
    int rowoff = (lane >> 4) * 8;              // lanes 16-31 hold M = r+8
    #pragma unroll
    for (int r = 0; r < 8; r++)
        cosm[(tm * 16 + rowoff + r) * NN + col] = c[r];
}

// ---------------- Kernel 3: top-20 indices per row (stable: lower idx wins ties)
__global__ void k_topk(const float* __restrict__ cosm, int* __restrict__ topk) {
    __shared__ float vals[NN];
    __shared__ float rv[256];
    __shared__ int   ri[256];
    int row = blockIdx.x;
    int t = threadIdx.x;
    for (int j = t; j < NN; j += 256) vals[j] = cosm[row * NN + j];
    __syncthreads();
    for (int p = 0; p < TOPK; p++) {
        float bv = -INFINITY; int bi = 0x7fffffff;
        for (int j = t; j < NN; j += 256) {
            float v = vals[j];
            if (v > bv || (v == bv && j < bi)) { bv = v; bi = j; }
        }
        rv[t] = bv; ri[t] = bi;
        __syncthreads();
        for (int s = 128; s > 0; s >>= 1) {
            if (t < s) {
                float v2 = rv[t + s]; int i2 = ri[t + s];
                if (v2 > rv[t] || (v2 == rv[t] && i2 < ri[t])) { rv[t] = v2; ri[t] = i2; }
            }
            __syncthreads();
        }
        if (t == 0) { topk[row * TOPK + p] = ri[0]; vals[ri[0]] = -INFINITY; }
        __syncthreads();
    }
}

// ---------------- Kernel 4: h = x @ W_lin via WMMA (K=10 padded to 12) -------
__global__ void k_h(const float* __restrict__ x, const float* __restrict__ W,
                    float* __restrict__ h) {
    int lane = threadIdx.x & 31;
    int wave = threadIdx.x >> 5;
    int t = blockIdx.x * 8 + wave;             // row tile 0..4095 (16 nodes each)
    int r16   = lane & 15;
    int khalf = (lane >> 4) * 2;
    int m = t * 16 + r16;
    #pragma unroll
    for (int tn = 0; tn < 4; tn++) {
        v8f c = {};
        #pragma unroll
        for (int k = 0; k < 12; k += 4) {
            int kk = k + khalf;
            int col = tn * 16 + r16;
            v2f a, b;
            a[0] = (kk     < FIN) ? x[m * FIN + kk]     : 0.f;
            a[1] = (kk + 1 < FIN) ? x[m * FIN + kk + 1] : 0.f;
            b[0] = (kk     < FIN) ? W[kk * DIM + col]       : 0.f;
            b[1] = (kk + 1 < FIN) ? W[(kk + 1) * DIM + col] : 0.f;
            c = __builtin_amdgcn_wmma_f32_16x16x4_f32(false, a, false, b, (short)0, c,
                                                      false, false);
        }
        int rowoff = (lane >> 4) * 8;
        #pragma unroll
        for (int r = 0; r < 8; r++)
            h[(t * 16 + rowoff + r) * DIM + tn * 16 + r16] = c[r];
    }
}

// ---------------- Kernel 5: per-node attention scalars sd, ss ----------------
__global__ void k_scores(const float* __restrict__ h, const float* __restrict__ emb,
                         const float* __restrict__ att_i, const float* __restrict__ att_j,
                         const float* __restrict__ att_em_i, const float* __restrict__ att_em_j,
                         float* __restrict__ sd, float* __restrict__ ss) {
    int lane = threadIdx.x & 31;
    int wave = threadIdx.x >> 5;
    int v = blockIdx.x * 8 + wave;             // node 0..65535
    int n = v & (NN - 1);
    float h0 = h[v * DIM + lane],     h1 = h[v * DIM + lane + 32];
    float e0 = emb[n * DIM + lane],   e1 = emb[n * DIM + lane + 32];
    float psd = h0 * att_i[lane] + h1 * att_i[lane + 32]
              + e0 * att_em_i[lane] + e1 * att_em_i[lane + 32];
    float pss = h0 * att_j[lane] + h1 * att_j[lane + 32]
              + e0 * att_em_j[lane] + e1 * att_em_j[lane + 32];
    psd = wave_sum(psd);
    pss = wave_sum(pss);
    if (lane == 0) { sd[v] = psd; ss[v] = pss; }
}

// ---------------- Kernel 6: softmax attention + aggregate + BN/ReLU head -----
// One wave per destination node. Lanes 0..19: top-k edges, lane 20: self loop.
__global__ void k_att(const float* __restrict__ h, const float* __restrict__ emb,
                      const int* __restrict__ topk, const float* __restrict__ sd,
                      const float* __restrict__ ss, const float* __restrict__ gnn_bias,
                      const float* __restrict__ g1, const float* __restrict__ be1,
                      const float* __restrict__ mu1, const float* __restrict__ va1,
                      const float* __restrict__ g2, const float* __restrict__ be2,
                      const float* __restrict__ mu2, const float* __restrict__ va2,
                      const float* __restrict__ outW, const float* __restrict__ outb,
                      float* __restrict__ out) {
    int lane = threadIdx.x & 31;
    int wave = threadIdx.x >> 5;
    int v = blockIdx.x * 8 + wave;             // dest node 0..65535
    int b = v >> 10;
    int i = v & (NN - 1);

    // per-lane edge logit (lanes 0..20)
    int jloc = i;
    bool valid = true;
    if (lane < TOPK) { jloc = topk[i * TOPK + lane]; valid = (jloc != i); } // removed self-loops
    int jg = (b << 10) + jloc;
    float logit = -INFINITY;
    if (lane <= TOPK && valid) {
        float l = sd[v] + ss[jg];
        logit = (l > 0.f) ? l : NEG_SLOPE * l;  // leaky_relu(0.2)
    }
    float mx = wave_max(logit);
    float ex = (lane <= TOPK && valid) ? __expf(logit - mx) : 0.f;
    float sum = wave_sum(ex);
    float alpha = ex / sum;

    // aggregate: each lane owns dims {lane, lane+32}
    float a0 = 0.f, a1 = 0.f;
    #pragma unroll
    for (int p = 0; p <= TOPK; p++) {
        float ap = __shfl(alpha, p, 32);       // wave-uniform
        int   jp = __shfl(jg,    p, 32);
        if (ap != 0.f) {
            a0 += ap * h[jp * DIM + lane];
            a1 += ap * h[jp * DIM + lane + 32];
        }
    }

    int d0 = lane, d1 = lane + 32;
    float x0 = a0 + gnn_bias[d0], x1 = a1 + gnn_bias[d1];
    float t0 = (x0 - mu1[d0]) * rsqrtf(va1[d0] + EPS) * g1[d0] + be1[d0];
    float t1 = (x1 - mu1[d1]) * rsqrtf(va1[d1] + EPS) * g1[d1] + be1[d1];
    t0 = fmaxf(t0, 0.f); t1 = fmaxf(t1, 0.f);
    float y0 = t0 * emb[i * DIM + d0];
    float y1 = t1 * emb[i * DIM + d1];
    float u0 = (y0 - mu2[d0]) * rsqrtf(va2[d0] + EPS) * g2[d0] + be2[d0];
    float u1 = (y1 - mu2[d1]) * rsqrtf(va2[d1] + EPS) * g2[d1] + be2[d1];
    float po = fmaxf(u0, 0.f) * outW[d0] + fmaxf(u1, 0.f) * outW[d1];
    po = wave_sum(po);
    if (lane == 0) out[v] = po + outb[0];
}

extern "C" void kernel_launch(void* const* d_in, const int* in_sizes, int n_in,
                              void* d_out, int out_size, void* d_ws, size_t ws_size,
                              hipStream_t stream) {
    const float* data     = (const float*)d_in[0];   // [B*N, FIN]
    const float* emb      = (const float*)d_in[1];   // [N, DIM]
    const float* W_lin    = (const float*)d_in[2];   // [FIN, DIM]
    const float* att_i    = (const float*)d_in[3];
    const float* att_j    = (const float*)d_in[4];
    const float* att_em_i = (const float*)d_in[5];
    const float* att_em_j = (const float*)d_in[6];
    const float* gnn_bias = (const float*)d_in[7];
    const float* bn1_g    = (const float*)d_in[8];
    const float* bn1_b    = (const float*)d_in[9];
    const float* bn1_m    = (const float*)d_in[10];
    const float* bn1_v    = (const float*)d_in[11];
    const float* bn2_g    = (const float*)d_in[12];
    const float* bn2_b    = (const float*)d_in[13];
    const float* bn2_m    = (const float*)d_in[14];
    const float* bn2_v    = (const float*)d_in[15];
    const float* out_W    = (const float*)d_in[16];
    const float* out_b    = (const float*)d_in[17];
    float* out = (float*)d_out;

    // workspace layout (bytes)
    char* ws = (char*)d_ws;
    size_t o = 0;
    float* wn   = (float*)(ws + o); o += (size_t)NN * DIM * 4;        // 256 KB
    float* cosm = (float*)(ws + o); o += (size_t)NN * NN * 4;         // 4 MB
    int*   topk = (int*)  (ws + o); o += (size_t)NN * TOPK * 4;       // 80 KB
    float* h    = (float*)(ws + o); o += (size_t)NODES * DIM * 4;     // 16 MB
    float* sd   = (float*)(ws + o); o += (size_t)NODES * 4;           // 256 KB
    float* ss   = (float*)(ws + o); o += (size_t)NODES * 4;           // 256 KB
    (void)ws_size; (void)in_sizes; (void)n_in; (void)out_size;

    k_normalize<<<NN, 32, 0, stream>>>(emb, wn);
    k_cos<<<(64 * 64) / 8, 256, 0, stream>>>(wn, cosm);
    k_topk<<<NN, 256, 0, stream>>>(cosm, topk);
    k_h<<<(NODES / 16) / 8, 256, 0, stream>>>(data, W_lin, h);
    k_scores<<<NODES / 8, 256, 0, stream>>>(h, emb, att_i, att_j, att_em_i, att_em_j,
                                            sd, ss);
    k_att<<<NODES / 8, 256, 0, stream>>>(h, emb, topk, sd, ss, gnn_bias,
                                         bn1_g, bn1_b, bn1_m, bn1_v,
                                         bn2_g, bn2_b, bn2_m, bn2_v,
                                         out_W, out_b, out);
}